// EpisodicSlotWriter_63823214018885
// MI455X (gfx1250) — compile-verified
//
#include <hip/hip_runtime.h>
#include <hip/hip_bf16.h>

typedef __attribute__((ext_vector_type(2))) float v2f;
typedef __attribute__((ext_vector_type(4))) float v4f;
typedef __attribute__((ext_vector_type(8))) float v8f;

// Problem sizes (fixed by the reference)
constexpr int NB = 1024;
constexpr int NK = 128;
constexpr int ND = 512;

// d_out layout: outputs concatenated flat in return order
constexpr size_t OFF_KEYS = 0;
constexpr size_t OFF_VALS = (size_t)NB * NK * ND;                 // 67,108,864
constexpr size_t OFF_AGE  = 2 * OFF_VALS;                         // 134,217,728
constexpr size_t OFF_STR  = OFF_AGE + (size_t)NB * NK;            // +131,072
constexpr size_t OFF_SLOT = OFF_STR + (size_t)NB * NK;
constexpr size_t OFF_BSIM = OFF_SLOT + (size_t)NB;

// ---------------------------------------------------------------------------
// Kernel 1: one block per batch row b. 256 threads = 8 waves; wave w owns
// keys [16w, 16w+16). Computes sim via chained V_WMMA_F32_16X16X4_F32,
// key norms as a VALU side product, then slot selection + age/strength
// outputs + per-row scalars for kernel 2. epi_keys loads are regular
// temporal (RT) so the rows stay resident in the 192 MB L2 for kernel 2.
// ---------------------------------------------------------------------------
__global__ __launch_bounds__(256) void epi_score_kernel(
    const float* __restrict__ write_key,
    const float* __restrict__ write_strength,
    const float* __restrict__ epi_keys,
    const float* __restrict__ epi_age,
    const float* __restrict__ epi_strength,
    float* __restrict__ out,
    int*   __restrict__ ws_slot,
    float* __restrict__ ws_alpha,
    float* __restrict__ ws_beta)
{
    const int b    = blockIdx.x;
    const int tid  = threadIdx.x;
    const int lane = tid & 31;
    const int wv   = tid >> 5;     // wave id 0..7 -> key tile
    const int mloc = lane & 15;    // M row within tile
    const int hi   = lane >> 4;    // half-wave (selects K pair 0/1 vs 2/3)

    __shared__ __align__(16) float s_wk[ND];   // query row (raw)
    __shared__ float s_red[256];               // ||wk||^2 reduction
    __shared__ float s_sqpart[256];            // per-lane partial ||ek||^2
    __shared__ float s_rawdot[NK];             // raw dot(ek, wk)
    __shared__ float s_v[NK];  __shared__ int s_i[NK];    // sim argmax
    __shared__ float s_av[NK]; __shared__ int s_ai[NK];   // age argmax
    __shared__ int   s_slot;
    __shared__ float s_ws;

    // --- load write_key row into LDS, reduce ||wk||^2 ---
    const float* wkp = write_key + (size_t)b * ND;
    float w0 = wkp[tid];
    float w1 = wkp[tid + 256];
    s_wk[tid]       = w0;
    s_wk[tid + 256] = w1;
    s_red[tid] = w0 * w0 + w1 * w1;
    __syncthreads();
    for (int s = 128; s > 0; s >>= 1) {
        if (tid < s) s_red[tid] += s_red[tid + s];
        __syncthreads();
    }

    // --- WMMA chain: rawdot for 16 keys per wave, contraction over D=512 ---
    // A (16x4 f32): lane m holds row (16*wv+m); VGPR0/1 = cols 4q+{0,1} for
    // lanes 0-15 and cols 4q+{2,3} for lanes 16-31  -> one v2f load per step.
    // B (4x16 f32): same K-pair selection, value broadcast over all N columns.
    const float* arow = epi_keys +
        ((size_t)b * NK + (size_t)(wv * 16 + mloc)) * ND + 2 * hi;
    const float* brow = s_wk + 2 * hi;

    v8f acc0 = {};
    v8f acc1 = {};
    float sq = 0.0f;
#pragma unroll 4
    for (int q = 0; q < 128; q += 2) {
        v2f a0 = *(const v2f*)(arow + 4 * q);
        v2f b0 = *(const v2f*)(brow + 4 * q);
        sq += a0.x * a0.x + a0.y * a0.y;
        acc0 = __builtin_amdgcn_wmma_f32_16x16x4_f32(
            false, a0, false, b0, (short)0, acc0, false, false);
        v2f a1 = *(const v2f*)(arow + 4 * q + 4);
        v2f b1 = *(const v2f*)(brow + 4 * q + 4);
        sq += a1.x * a1.x + a1.y * a1.y;
        acc1 = __builtin_amdgcn_wmma_f32_16x16x4_f32(
            false, a1, false, b1, (short)0, acc1, false, false);
    }
    v8f acc = acc0 + acc1;

    s_sqpart[tid] = sq;   // covers cols ==0,1 mod 4 (hi=0) / ==2,3 mod 4 (hi=1)
    // D[M][N]: VGPR v -> M = v + 8*hi, all N columns identical (B broadcast).
    if (mloc == 0) {
#pragma unroll
        for (int v = 0; v < 8; ++v)
            s_rawdot[wv * 16 + hi * 8 + v] = acc[v];
    }
    __syncthreads();

    // --- per-key sim + age score ---
    if (tid < NK) {
        const int w2 = tid >> 4, m2 = tid & 15;
        float sqn = s_sqpart[w2 * 32 + m2] + s_sqpart[w2 * 32 + m2 + 16];
        float wknorm = sqrtf(s_red[0]);
        float sim = s_rawdot[tid] / ((sqrtf(sqn) + 1e-6f) * (wknorm + 1e-6f));
        s_v[tid] = sim;  s_i[tid] = tid;
        float age = epi_age[(size_t)b * NK + tid];
        float st  = epi_strength[(size_t)b * NK + tid];
        float stc = fminf(fmaxf(st, 0.0f), 1.0f);
        s_av[tid] = age + (1.0f - stc) * 0.01f;  s_ai[tid] = tid;
    }
    __syncthreads();

    // --- argmax reductions (strict > keeps first occurrence, like jnp.argmax)
    for (int s = 64; s > 0; s >>= 1) {
        if (tid < s) {
            if (s_v[tid + s] > s_v[tid])  { s_v[tid] = s_v[tid + s];  s_i[tid] = s_i[tid + s]; }
            if (s_av[tid + s] > s_av[tid]){ s_av[tid] = s_av[tid + s]; s_ai[tid] = s_ai[tid + s]; }
        }
        __syncthreads();
    }

    if (tid == 0) {
        float bsim = s_v[0];
        int slot = (bsim > 0.85f) ? s_i[0] : s_ai[0];
        float wsv = fminf(fmaxf(write_strength[b], 0.0f), 1.0f);
        s_slot = slot;
        s_ws   = wsv;
        out[OFF_SLOT + b] = (float)slot;
        out[OFF_BSIM + b] = bsim;
        ws_slot[b]  = slot;
        ws_alpha[b] = 0.25f * wsv;
        ws_beta[b]  = 0.25f * wsv;
    }
    __syncthreads();

    // --- age / strength outputs ---
    if (tid < NK) {
        const int slot = s_slot;
        const float wsv = s_ws;
        float age = epi_age[(size_t)b * NK + tid];
        float st  = epi_strength[(size_t)b * NK + tid] * 0.999f;  // decay
        float agen = (tid == slot) ? 0.0f : (age + 1.0f);
        float stn = st;
        if (tid == slot) {
            stn = st + wsv * (1.0f - st);
            stn = fminf(fmaxf(stn, 0.001f), 1.0f);
        }
        out[OFF_AGE + (size_t)b * NK + tid] = agen;
        out[OFF_STR + (size_t)b * NK + tid] = stn;
    }
}

// ---------------------------------------------------------------------------
// Kernel 2: streaming copy of epi_keys/epi_vals to the outputs (B128 per
// thread), with the single (b, k==slot) block doing the blend + renormalize.
// One block per (b,k) row: 131072 blocks x 128 threads, 1 float4 per array.
// All streaming traffic is marked non-temporal (TH=NT): the 537 MB of output
// stores must not evict the epi_keys rows kernel 1 just parked in L2, and
// epi_vals/epi_keys reads here are single-use/last-use.
// ---------------------------------------------------------------------------
__global__ __launch_bounds__(128) void epi_copy_kernel(
    const float* __restrict__ write_key,
    const float* __restrict__ write_val,
    const float* __restrict__ epi_keys,
    const float* __restrict__ epi_vals,
    const int*   __restrict__ ws_slot,
    const float* __restrict__ ws_alpha,
    const float* __restrict__ ws_beta,
    float* __restrict__ out)
{
    const int bid = blockIdx.x;
    const int b = bid >> 7;      // / NK
    const int k = bid & (NK - 1);
    const int t = threadIdx.x;

    const size_t rowoff = ((size_t)b * NK + k) * ND;
    const v4f* kin = (const v4f*)(epi_keys + rowoff);
    const v4f* vin = (const v4f*)(epi_vals + rowoff);
    v4f* kout = (v4f*)(out + OFF_KEYS + rowoff);
    v4f* vout = (v4f*)(out + OFF_VALS + rowoff);

    // keys: last use (kernel 1 already consumed them); vals: only use.
    v4f kv = __builtin_nontemporal_load(kin + t);
    v4f vv = __builtin_nontemporal_load(vin + t);
    const int slot = ws_slot[b];   // uniform per block

    if (k != slot) {
        __builtin_nontemporal_store(kv, kout + t);
        __builtin_nontemporal_store(vv, vout + t);
    } else {
        __shared__ float s_n[128];
        const float a  = ws_alpha[b];
        const float be = ws_beta[b];
        const v4f* wkp = (const v4f*)(write_key + (size_t)b * ND);
        const v4f* wvp = (const v4f*)(write_val + (size_t)b * ND);
        v4f wk4 = wkp[t];
        v4f wv4 = wvp[t];

        v4f nk = (1.0f - a) * kv + a * wk4;
        v4f nv = (1.0f - be) * vv + be * wv4;

        s_n[t] = nk.x * nk.x + nk.y * nk.y + nk.z * nk.z + nk.w * nk.w;
        __syncthreads();
        for (int s = 64; s > 0; s >>= 1) {
            if (t < s) s_n[t] += s_n[t + s];
            __syncthreads();
        }
        const float inv = 1.0f / (sqrtf(s_n[0]) + 1e-6f);
        nk *= inv;

        __builtin_nontemporal_store(nk, kout + t);
        __builtin_nontemporal_store(nv, vout + t);
    }
}

extern "C" void kernel_launch(void* const* d_in, const int* in_sizes, int n_in,
                              void* d_out, int out_size, void* d_ws, size_t ws_size,
                              hipStream_t stream) {
    const float* write_key      = (const float*)d_in[0];
    const float* write_val      = (const float*)d_in[1];
    const float* write_strength = (const float*)d_in[2];
    const float* epi_keys       = (const float*)d_in[3];
    const float* epi_vals       = (const float*)d_in[4];
    const float* epi_age        = (const float*)d_in[5];
    const float* epi_strength   = (const float*)d_in[6];
    float* out = (float*)d_out;

    // workspace: [NB ints slot][NB floats alpha][NB floats beta]
    int*   ws_slot  = (int*)d_ws;
    float* wsf      = (float*)d_ws;
    float* ws_alpha = wsf + NB;
    float* ws_beta  = wsf + 2 * NB;

    epi_score_kernel<<<NB, 256, 0, stream>>>(
        write_key, write_strength, epi_keys, epi_age, epi_strength,
        out, ws_slot, ws_alpha, ws_beta);

    epi_copy_kernel<<<NB * NK, 128, 0, stream>>>(
        write_key, write_val, epi_keys, epi_vals,
        ws_slot, ws_alpha, ws_beta, out);
}